// MeToken_24627342475478
// MI455X (gfx1250) — compile-verified
//
#include <hip/hip_runtime.h>
#include <math.h>

typedef __attribute__((ext_vector_type(2))) float v2f;
typedef __attribute__((ext_vector_type(8))) float v8f;

#define DIMD   512
#define NPT    128       // codes per type
#define NTYPES 26
#define TB     64        // tokens per block (4 sub-tiles of 16)
#define NST    4         // sub-tiles
#define LSTR   516       // padded LDS row stride (floats) -> conflict-free frag reads
#define EPSN   1e-12f

// ---------------- helper kernels ----------------

__global__ void zero_kernel(int* counts, int* cursors) {
    int tid = threadIdx.x;
    if (tid < 32) counts[tid] = 0;
    else if (tid < 64) cursors[tid - 32] = 0;
}

__global__ void count_kernel(const int* __restrict__ Q, int* __restrict__ counts, int N) {
    int n = blockIdx.x * blockDim.x + threadIdx.x;
    if (n < N) atomicAdd(&counts[Q[n]], 1);
}

__global__ void scan_kernel(const int* __restrict__ counts, int* __restrict__ offsets,
                            int* __restrict__ cursors) {
    if (threadIdx.x == 0) {
        int run = 0;
        for (int t = 0; t < NTYPES; ++t) {
            offsets[t] = run; cursors[t] = run; run += counts[t];
        }
    }
}

__global__ void scatter_kernel(const int* __restrict__ Q, int* __restrict__ cursors,
                               int* __restrict__ sorted, int N) {
    int n = blockIdx.x * blockDim.x + threadIdx.x;
    if (n < N) {
        int pos = atomicAdd(&cursors[Q[n]], 1);
        sorted[pos] = n;
    }
}

__global__ __launch_bounds__(128) void esq_kernel(const float* __restrict__ emb,
                                                  float* __restrict__ esq) {
    int row = blockIdx.x;
    int tid = threadIdx.x;
    const float* er = emb + (size_t)row * DIMD;
    float s = 0.0f;
    for (int c = tid; c < DIMD; c += 128) { float v = er[c]; s += v * v; }
    __shared__ float red[128];
    red[tid] = s; __syncthreads();
    for (int st = 64; st > 0; st >>= 1) {
        if (tid < st) red[tid] += red[tid + st];
        __syncthreads();
    }
    if (tid == 0) esq[row] = red[0];
}

// ---------------- main fused VQ kernel (WMMA f32, 64-token tile) ----------------

__global__ __launch_bounds__(128) void vq_main_kernel(
    const float* __restrict__ x,
    const float* __restrict__ emb,
    const int*   __restrict__ sorted,
    const int*   __restrict__ offsets,
    const int*   __restrict__ counts,
    const float* __restrict__ esq,
    float* __restrict__ outQ,       // [N, 512] quantized_st
    float* __restrict__ outIdx,     // [N] encoding index (as float)
    float* __restrict__ tokenDiff)  // [N] per-token sum of (q - xn)^2
{
    const int tid  = threadIdx.x;
    const int t    = blockIdx.y;          // type
    const int tile = blockIdx.x;
    const int off  = offsets[t];
    const int cnt  = counts[t];
    const int start = tile * TB;
    if (start >= cnt) return;             // empty tile, whole block exits
    const int count64 = min(TB, cnt - start);

    // CDNA5 allows up to 320KB LDS per workgroup; we use ~146KB.
    __shared__ float ldsA[TB * LSTR];     // normalized x tile (64 x 512, padded)
    __shared__ int   tokenIds[TB];
    __shared__ float normRed[128];
    __shared__ float invn[TB];
    __shared__ float minVal[TB * 8];
    __shared__ int   minIdx[TB * 8];
    __shared__ int   bestEnc[TB];
    __shared__ float invq[TB];
    __shared__ float partLDS[16 * 132];

    if (tid < TB) tokenIds[tid] = (tid < count64) ? sorted[off + start + tid] : -1;
    __syncthreads();

    // load 64x512 tile (zeros for padding rows); coalesced
    for (int idx = tid; idx < TB * DIMD; idx += 128) {
        int r = idx >> 9, c = idx & (DIMD - 1);
        float v = 0.0f;
        if (r < count64) v = x[(size_t)tokenIds[r] * DIMD + c];
        ldsA[r * LSTR + c] = v;
    }
    __syncthreads();

    // row norms: 2 threads per row, 256 elems each
    {
        int r = tid >> 1, sub = tid & 1;
        float s = 0.0f;
        const float* row = &ldsA[r * LSTR + sub * 256];
        for (int i = 0; i < 256; ++i) { float v = row[i]; s += v * v; }
        normRed[tid] = s;
    }
    __syncthreads();
    if (tid < TB) {
        float s = normRed[tid * 2] + normRed[tid * 2 + 1];
        invn[tid] = 1.0f / fmaxf(sqrtf(s), EPSN);
    }
    __syncthreads();
    for (int idx = tid; idx < TB * DIMD; idx += 128) {
        int r = idx >> 9;
        ldsA[r * LSTR + (idx & (DIMD - 1))] *= invn[r];
    }
    __syncthreads();

    // WMMA fp32 GEMM: each wave owns two 16-code tiles; each B fragment is
    // reused across the 4 token sub-tiles (4 accumulators).
    const int lane = tid & 31;
    const int wave = tid >> 5;
    const int lo = lane & 15;
    const int hi = lane >> 4;
    const float* arow0 = &ldsA[(0 * 16 + lo) * LSTR + 2 * hi];
    const float* arow1 = &ldsA[(1 * 16 + lo) * LSTR + 2 * hi];
    const float* arow2 = &ldsA[(2 * 16 + lo) * LSTR + 2 * hi];
    const float* arow3 = &ldsA[(3 * 16 + lo) * LSTR + 2 * hi];

    for (int w2 = 0; w2 < 2; ++w2) {
        const int ct = wave * 2 + w2;
        const int codeRow = t * NPT + ct * 16 + lo;
        const float* brow = emb + (size_t)codeRow * DIMD + 2 * hi;
        v8f acc0 = {0.f,0.f,0.f,0.f,0.f,0.f,0.f,0.f};
        v8f acc1 = acc0, acc2 = acc0, acc3 = acc0;
        #pragma unroll 4
        for (int kk = 0; kk < DIMD; kk += 4) {
            v2f b = *(const v2f*)(brow + kk);
            v2f a0 = *(const v2f*)(arow0 + kk);
            v2f a1 = *(const v2f*)(arow1 + kk);
            v2f a2 = *(const v2f*)(arow2 + kk);
            v2f a3 = *(const v2f*)(arow3 + kk);
            acc0 = __builtin_amdgcn_wmma_f32_16x16x4_f32(false, a0, false, b, (short)0, acc0, false, false);
            acc1 = __builtin_amdgcn_wmma_f32_16x16x4_f32(false, a1, false, b, (short)0, acc1, false, false);
            acc2 = __builtin_amdgcn_wmma_f32_16x16x4_f32(false, a2, false, b, (short)0, acc2, false, false);
            acc3 = __builtin_amdgcn_wmma_f32_16x16x4_f32(false, a3, false, b, (short)0, acc3, false, false);
        }
        const float es = esq[codeRow];
        #pragma unroll
        for (int st = 0; st < NST; ++st) {
            const v8f acc = (st == 0) ? acc0 : (st == 1) ? acc1 : (st == 2) ? acc2 : acc3;
            #pragma unroll
            for (int r = 0; r < 8; ++r) {
                float v = es - 2.0f * acc[r];   // dist minus constant ||xn||^2
                int ci = ct * 16 + lo;
                #pragma unroll
                for (int m = 8; m >= 1; m >>= 1) {
                    float ov = __shfl_xor(v, m, 16);
                    int   oi = __shfl_xor(ci, m, 16);
                    if (ov < v || (ov == v && oi < ci)) { v = ov; ci = oi; }
                }
                if (lo == 0) {
                    int token = st * 16 + r + hi * 8;
                    minVal[token * 8 + ct] = v;
                    minIdx[token * 8 + ct] = ci;
                }
            }
        }
    }
    __syncthreads();

    // cross-tile argmin (lowest index wins ties, ct ascending = index ascending)
    if (tid < TB) {
        float bv = minVal[tid * 8];
        int   bi = minIdx[tid * 8];
        for (int ct = 1; ct < 8; ++ct) {
            float v = minVal[tid * 8 + ct];
            int  i2 = minIdx[tid * 8 + ct];
            if (v < bv || (v == bv && i2 < bi)) { bv = v; bi = i2; }
        }
        int enc = t * NPT + bi;
        bestEnc[tid] = enc;
        invq[tid] = 1.0f / fmaxf(sqrtf(esq[enc]), EPSN);
        if (tid < count64) outIdx[tokenIds[tid]] = (float)enc;
    }
    __syncthreads();

    // quantized_st + per-token diff^2, processed in 4 groups of 16 rows to
    // keep reduction scratch small; coalesced and deterministic per token
    for (int g = 0; g < 4; ++g) {
        for (int rr = 0; rr < 16; ++rr) {
            const int r = g * 16 + rr;
            float part = 0.0f;
            if (r < count64) {
                const float* er = emb + (size_t)bestEnc[r] * DIMD;
                const float iq = invq[r];
                float* orow = outQ + (size_t)tokenIds[r] * DIMD;
                #pragma unroll
                for (int k = 0; k < 4; ++k) {
                    int c = k * 128 + tid;
                    float xv = ldsA[r * LSTR + c];
                    float q  = er[c] * iq;
                    float dd = q - xv;
                    orow[c] = xv + dd;          // xn + (quantized - xn), as in reference
                    part += dd * dd;
                }
            }
            partLDS[rr * 132 + tid] = part;
        }
        __syncthreads();
        if (tid < 16) {
            const int r = g * 16 + tid;
            if (r < count64) {
                float s = 0.0f;
                for (int k = 0; k < 128; ++k) s += partLDS[tid * 132 + k];
                tokenDiff[tokenIds[r]] = s;
            }
        }
        __syncthreads();
    }
}

// ---------------- uniform (contrastive) loss ----------------

__global__ __launch_bounds__(128) void uniform_kernel(
    const float* __restrict__ emb,
    const int*   __restrict__ sampled,
    const float* __restrict__ esq,
    float* __restrict__ uniPart, int S)
{
    const int i = blockIdx.x;
    const int tid = threadIdx.x;
    const int icode = sampled[i];
    const int labi = icode >> 7;  // / NUM_PER_TYPE
    __shared__ float sei[DIMD];
    const float invi = 1.0f / fmaxf(sqrtf(esq[icode]), EPSN);
    for (int c = tid; c < DIMD; c += 128) sei[c] = emb[(size_t)icode * DIMD + c] * invi;
    __syncthreads();

    float se = 0.0f, sp = 0.0f;
    for (int j = tid; j < S; j += 128) {
        if (j == i) continue;
        int jc = sampled[j];
        const float* er = emb + (size_t)jc * DIMD;
        float d = 0.0f;
        for (int c = 0; c < DIMD; ++c) d += sei[c] * er[c];
        d *= 1.0f / fmaxf(sqrtf(esq[jc]), EPSN);
        float e = expf(d / 0.07f);
        se += e;
        if ((jc >> 7) == labi) sp += e;
    }
    __shared__ float rs[128], rp[128];
    rs[tid] = se; rp[tid] = sp; __syncthreads();
    for (int s = 64; s > 0; s >>= 1) {
        if (tid < s) { rs[tid] += rs[tid + s]; rp[tid] += rp[tid + s]; }
        __syncthreads();
    }
    if (tid == 0) uniPart[i] = logf(rs[0]) - logf(rp[0]);  // -log(pos/sum)
}

// ---------------- deterministic final reductions ----------------

__global__ __launch_bounds__(256) void finalize_kernel(
    const float* __restrict__ tokenDiff,
    const float* __restrict__ uniPart,
    float* __restrict__ outScalars, int N, int S)
{
    const int tid = threadIdx.x;
    __shared__ float red[256];
    const int chunk = N / 256;  // fixed contiguous chunks -> deterministic order
    float s = 0.0f;
    for (int k = 0; k < chunk; ++k) s += tokenDiff[tid * chunk + k];
    red[tid] = s;
    __syncthreads();
    if (tid == 0) {
        float tot = 0.0f;
        for (int k = 0; k < 256; ++k) tot += red[k];
        // loss = q_latent + 0.25*e_latent, both == mean((q-xn)^2) in fwd
        outScalars[0] = tot * 1.25f / ((float)N * (float)DIMD);
        float u = 0.0f;
        for (int k = 0; k < S; ++k) u += uniPart[k];
        outScalars[1] = u / (float)S;
    }
}

// ---------------- launcher ----------------

extern "C" void kernel_launch(void* const* d_in, const int* in_sizes, int n_in,
                              void* d_out, int out_size, void* d_ws, size_t ws_size,
                              hipStream_t stream) {
    (void)n_in; (void)out_size; (void)ws_size;
    const float* x       = (const float*)d_in[0];
    const int*   Q       = (const int*)d_in[1];
    const float* emb     = (const float*)d_in[2];
    const int*   sampled = (const int*)d_in[3];
    const int N = in_sizes[1];
    const int S = in_sizes[3];
    const int NCODES = in_sizes[2] / DIMD;   // 3328
    float* out = (float*)d_out;
    const size_t ND = (size_t)N * DIMD;

    // workspace layout (all 4B-aligned)
    int*   counts    = (int*)d_ws;           // 32
    int*   offsets   = counts + 32;          // 32
    int*   cursors   = offsets + 32;         // 32
    int*   sorted    = cursors + 32;         // N
    float* esq       = (float*)(sorted + N); // NCODES
    float* tokenDiff = esq + NCODES;         // N
    float* uniPart   = tokenDiff + N;        // S

    zero_kernel   <<<1, 64, 0, stream>>>(counts, cursors);
    count_kernel  <<<(N + 255) / 256, 256, 0, stream>>>(Q, counts, N);
    scan_kernel   <<<1, 32, 0, stream>>>(counts, offsets, cursors);
    scatter_kernel<<<(N + 255) / 256, 256, 0, stream>>>(Q, cursors, sorted, N);
    esq_kernel    <<<NCODES, 128, 0, stream>>>(emb, esq);

    dim3 grid((N + TB - 1) / TB, NTYPES);
    vq_main_kernel<<<grid, 128, 0, stream>>>(x, emb, sorted, offsets, counts, esq,
                                             out, out + ND + 2, tokenDiff);

    uniform_kernel <<<S, 128, 0, stream>>>(emb, sampled, esq, uniPart, S);
    finalize_kernel<<<1, 256, 0, stream>>>(tokenDiff, uniPart, out + ND, N, S);
}